// LabelSmoothingOrdinalLoss_9861244911667
// MI455X (gfx1250) — compile-verified
//
#include <hip/hip_runtime.h>

// ---------------------------------------------------------------------------
// LabelSmoothingOrdinalLoss for MI455X (gfx1250, wave32).
// Memory-bound streaming reduction (~176 MiB read once -> scalar), floor
// ~7.9us at 23.3 TB/s. NT loads to avoid L2 rinse; per-sample dot computed
// as a 16x16 GEMM via chained V_WMMA_F32_16X16X4_F32 (K=10 padded to 12);
// deterministic two-stage reduction through d_ws (no float atomics).
// ---------------------------------------------------------------------------

#define NCLS 10

typedef __attribute__((ext_vector_type(2))) float v2f;
typedef __attribute__((ext_vector_type(8))) float v8f;

struct SmoothTable { float v[16][16]; };   // padded 16x16, pad = 0

// Exact replication of the Python float64 table construction, f32 on write.
constexpr SmoothTable make_table() {
  SmoothTable T{};  // zero-init (padding rows/cols stay 0)
  for (int t = 0; t < NCLS; ++t) {
    T.v[t][t] = (float)(1.0 - 0.1);
    double rem = 0.1;
    for (int d = 1; d < NCLS; ++d) {
      double w = 1.0 - (double)d * 0.3;
      if (w < 0.0) w = 0.0;
      for (int dir = -1; dir <= 1; dir += 2) {     // -1 first, then +1
        int adj = t + dir * d;
        if (0 <= adj && adj < NCLS) {
          double p = rem * w / 2.0;
          T.v[t][adj] = (float)p;
          rem -= p;
        }
      }
      if (rem <= 0.0) break;
    }
  }
  return T;
}

__constant__ const SmoothTable g_tbl = make_table();

__global__ __launch_bounds__(256)
void loss_main(const float* __restrict__ pred, const int* __restrict__ tgt,
               float* __restrict__ partial, int n) {
  // LDS staging. lp stride 12 floats: 12*r mod 64 hits 16 distinct banks ->
  // conflict-free A fetches. dt stride 20 floats: lane halves land on
  // disjoint bank ranges for the D scatter.
  __shared__ float lp[256][12];
  __shared__ float dt[256][20];
  __shared__ float red[256];

  const int tid   = threadIdx.x;
  const int lane  = tid & 31;
  const int wbase = tid & ~31;        // this wave's first LDS row
  const int half  = lane >> 4;        // 0: lanes 0-15, 1: lanes 16-31
  const int mrow  = lane & 15;

  // Zero the padded class columns once (never written again).
  lp[tid][10] = 0.0f;
  lp[tid][11] = 0.0f;

  // Loop-invariant B operands for D = logp * table^T.
  // B element (K=c_local, N=t): lane holds N=mrow, VGPR r holds K = r + 2*half.
  v2f b0, b1, b2;
  b0.x = g_tbl.v[mrow][0 + 2 * half];  b0.y = g_tbl.v[mrow][1 + 2 * half];
  b1.x = g_tbl.v[mrow][4 + 2 * half];  b1.y = g_tbl.v[mrow][5 + 2 * half];
  b2.x = g_tbl.v[mrow][8 + 2 * half];  b2.y = g_tbl.v[mrow][9 + 2 * half];

  const long long stride = (long long)gridDim.x * blockDim.x;
  const long long base   = (long long)blockIdx.x * blockDim.x + tid;
  const long long iters  = (long long)n / stride;   // uniform across the grid
  float accf = 0.0f;

  for (long long it = 0; it < iters; ++it) {
    const long long s = base + it * stride;

    // 40B/sample: five 8B-aligned NT b64 loads (streamed once -> TH=NT).
    const v2f* p = (const v2f*)(pred + s * 10);
    v2f x0 = __builtin_nontemporal_load(p + 0);
    v2f x1 = __builtin_nontemporal_load(p + 1);
    v2f x2 = __builtin_nontemporal_load(p + 2);
    v2f x3 = __builtin_nontemporal_load(p + 3);
    v2f x4 = __builtin_nontemporal_load(p + 4);
    const int t = __builtin_nontemporal_load(tgt + s);

    // Stable log-softmax in registers.
    float m = fmaxf(fmaxf(fmaxf(fmaxf(x0.x, x0.y), fmaxf(x1.x, x1.y)),
                          fmaxf(fmaxf(x2.x, x2.y), fmaxf(x3.x, x3.y))),
                    fmaxf(x4.x, x4.y));
    float sum = __expf(x0.x - m) + __expf(x0.y - m) +
                __expf(x1.x - m) + __expf(x1.y - m) +
                __expf(x2.x - m) + __expf(x2.y - m) +
                __expf(x3.x - m) + __expf(x3.y - m) +
                __expf(x4.x - m) + __expf(x4.y - m);
    const float lse = __logf(sum) + m;

    // Stage logp into this thread's LDS row.
    float* row = lp[tid];
    v2f l;
    l.x = x0.x - lse; l.y = x0.y - lse; *(v2f*)&row[0] = l;
    l.x = x1.x - lse; l.y = x1.y - lse; *(v2f*)&row[2] = l;
    l.x = x2.x - lse; l.y = x2.y - lse; *(v2f*)&row[4] = l;
    l.x = x3.x - lse; l.y = x3.y - lse; *(v2f*)&row[6] = l;
    l.x = x4.x - lse; l.y = x4.y - lse; *(v2f*)&row[8] = l;
    __builtin_amdgcn_wave_barrier();   // LDS is in-order per wave; pin codegen

    // Two 16-sample tiles per wave; D[s,k] = sum_c logp[s,c] * table[k,c].
#pragma unroll
    for (int tile = 0; tile < 2; ++tile) {
      const float* arow = lp[wbase + 16 * tile + mrow];   // A: M = lane%16
      v2f a0 = *(const v2f*)&arow[0 + 2 * half];          // K = 0..3
      v2f a1 = *(const v2f*)&arow[4 + 2 * half];          // K = 4..7
      v2f a2 = *(const v2f*)&arow[8 + 2 * half];          // K = 8..11 (pad 0)
      v8f c = {};
      c = __builtin_amdgcn_wmma_f32_16x16x4_f32(false, a0, false, b0, (short)0, c, false, false);
      c = __builtin_amdgcn_wmma_f32_16x16x4_f32(false, a1, false, b1, (short)0, c, false, false);
      c = __builtin_amdgcn_wmma_f32_16x16x4_f32(false, a2, false, b2, (short)0, c, false, false);
      // Scatter D: VGPR r, this lane -> (M = r + 8*half, N = mrow).
#pragma unroll
      for (int r = 0; r < 8; ++r)
        dt[wbase + 16 * tile + 8 * half + r][mrow] = c[r];
    }
    __builtin_amdgcn_wave_barrier();

    // Gather my sample's column t: loss_s = -D[s, t].
    accf += dt[tid][t];
    __builtin_amdgcn_wave_barrier();   // before next iter reuses lp/dt
  }

  // Scalar VALU tail (empty when stride | n, e.g. n = 4Mi). No WMMA here,
  // so divergence is safe.
  for (long long s = base + iters * stride; s < n; s += stride) {
    const v2f* p = (const v2f*)(pred + s * 10);
    v2f x0 = p[0], x1 = p[1], x2 = p[2], x3 = p[3], x4 = p[4];
    float m = fmaxf(fmaxf(fmaxf(fmaxf(x0.x, x0.y), fmaxf(x1.x, x1.y)),
                          fmaxf(fmaxf(x2.x, x2.y), fmaxf(x3.x, x3.y))),
                    fmaxf(x4.x, x4.y));
    float sum = __expf(x0.x - m) + __expf(x0.y - m) +
                __expf(x1.x - m) + __expf(x1.y - m) +
                __expf(x2.x - m) + __expf(x2.y - m) +
                __expf(x3.x - m) + __expf(x3.y - m) +
                __expf(x4.x - m) + __expf(x4.y - m);
    const float lse = __logf(sum) + m;
    const int t = tgt[s];
    const float* tr = g_tbl.v[t];
    float dot = tr[0] * (x0.x - lse) + tr[1] * (x0.y - lse) +
                tr[2] * (x1.x - lse) + tr[3] * (x1.y - lse) +
                tr[4] * (x2.x - lse) + tr[5] * (x2.y - lse) +
                tr[6] * (x3.x - lse) + tr[7] * (x3.y - lse) +
                tr[8] * (x4.x - lse) + tr[9] * (x4.y - lse);
    accf += dot;
  }

  // Deterministic block reduction -> per-block partial.
  red[tid] = accf;
  __syncthreads();
  for (int off = 128; off > 0; off >>= 1) {
    if (tid < off) red[tid] += red[tid + off];
    __syncthreads();
  }
  if (tid == 0) partial[blockIdx.x] = red[0];
}

__global__ __launch_bounds__(256)
void loss_finish(const float* __restrict__ partial, int nblocks, int n,
                 float* __restrict__ out) {
  __shared__ double red[256];
  double a = 0.0;
  for (int i = threadIdx.x; i < nblocks; i += 256) a += (double)partial[i];
  red[threadIdx.x] = a;
  __syncthreads();
  for (int off = 128; off > 0; off >>= 1) {
    if (threadIdx.x < off) red[threadIdx.x] += red[threadIdx.x + off];
    __syncthreads();
  }
  if (threadIdx.x == 0) out[0] = (float)(-red[0] / (double)n);
}

extern "C" void kernel_launch(void* const* d_in, const int* in_sizes, int n_in,
                              void* d_out, int out_size, void* d_ws, size_t ws_size,
                              hipStream_t stream) {
  const float* pred = (const float*)d_in[0];   // [B, 10] f32
  const int*   tgt  = (const int*)d_in[1];     // [B] i32
  float* out     = (float*)d_out;              // scalar f32
  float* partial = (float*)d_ws;

  const int n = in_sizes[1];                   // batch size (4,194,304)

  int blocks = 4096;                           // 1Mi threads -> 4 samples/thread
  long long total = (long long)blocks * 256;
  if ((long long)n < total) blocks = (n + 255) / 256;
  if ((size_t)blocks * sizeof(float) > ws_size && ws_size >= sizeof(float))
    blocks = (int)(ws_size / sizeof(float));
  if (blocks < 1) blocks = 1;

  loss_main<<<blocks, 256, 0, stream>>>(pred, tgt, partial, n);
  loss_finish<<<1, 256, 0, stream>>>(partial, blocks, n, out);
}